// GCNLayer_38500086841975
// MI455X (gfx1250) — compile-verified
//
#include <hip/hip_runtime.h>
#include <hip/hip_bf16.h>

typedef __attribute__((ext_vector_type(2))) float v2f;
typedef __attribute__((ext_vector_type(8))) float v8f;

#define D128 128
#define ALPHA 0.2f

static __device__ __forceinline__ void atomAdd(float* p, float v) {
    __hip_atomic_fetch_add(p, v, __ATOMIC_RELAXED, __HIP_MEMORY_SCOPE_AGENT);
}

// ---------------------------------------------------------------------------
// K0: x0 = tanh(x); h = x0; zero agg128, agg256, norm_o, norm_i
// grid covers N*256 threads
// ---------------------------------------------------------------------------
__global__ void k_init(const float* __restrict__ x, float* __restrict__ x0,
                       float* __restrict__ h, float* __restrict__ agg128,
                       float* __restrict__ agg256, float* __restrict__ no,
                       float* __restrict__ ni, int N) {
    size_t idx = (size_t)blockIdx.x * blockDim.x + threadIdx.x;
    size_t n128 = (size_t)N * D128;
    size_t n256 = (size_t)N * 2 * D128;
    if (idx < n128) {
        float t = tanhf(x[idx]);
        x0[idx] = t;
        h[idx]  = t;
        agg128[idx] = 0.f;
    }
    if (idx < n256) agg256[idx] = 0.f;
    if (idx < (size_t)N) { no[idx] = 0.f; ni[idx] = 0.f; }
}

// ---------------------------------------------------------------------------
// K1: accumulate degrees (as floats) with hardware f32 atomics
// ---------------------------------------------------------------------------
__global__ void k_degrees(const int* __restrict__ src, const int* __restrict__ dst,
                          float* __restrict__ no, float* __restrict__ ni, int E) {
    int e = blockIdx.x * blockDim.x + threadIdx.x;
    if (e >= E) return;
    atomAdd(no + src[e], 1.0f);
    atomAdd(ni + dst[e], 1.0f);
}

// K2: norm = rsqrt(max(deg,1)) in place
__global__ void k_norm_fin(float* __restrict__ no, float* __restrict__ ni, int N) {
    int i = blockIdx.x * blockDim.x + threadIdx.x;
    if (i >= N) return;
    no[i] = rsqrtf(fmaxf(no[i], 1.0f));
    ni[i] = rsqrtf(fmaxf(ni[i], 1.0f));
}

// ---------------------------------------------------------------------------
// K3: APPNP scatter: agg[dst] += h[src] * norm_o[src]
// one wave (32 lanes) per edge, float4 per lane -> global_load_b128 gathers
// ---------------------------------------------------------------------------
__global__ void k_scatter128_scaled(const float* __restrict__ h,
                                    const float* __restrict__ no,
                                    const int* __restrict__ src,
                                    const int* __restrict__ dst,
                                    float* __restrict__ agg, int E) {
    size_t t = (size_t)blockIdx.x * blockDim.x + threadIdx.x;
    int e = (int)(t >> 5);
    if (e >= E) return;
    int lane = (int)(t & 31);
    int s = src[e], d = dst[e];
    float ns = no[s];
    const float4 v = *reinterpret_cast<const float4*>(h + (size_t)s * D128 + lane * 4);
    float* ap = agg + (size_t)d * D128 + lane * 4;
    atomAdd(ap + 0, v.x * ns);
    atomAdd(ap + 1, v.y * ns);
    atomAdd(ap + 2, v.z * ns);
    atomAdd(ap + 3, v.w * ns);
}

// K4: h = (1-a)*agg*norm_i + a*x0 ; agg = 0 (ready for next round)
__global__ void k_appnp_update(float* __restrict__ agg, const float* __restrict__ x0,
                               const float* __restrict__ ni, float* __restrict__ h, int N) {
    size_t idx = (size_t)blockIdx.x * blockDim.x + threadIdx.x;
    if (idx >= (size_t)N * D128) return;
    size_t i = idx >> 7;
    float a = agg[idx];
    h[idx] = (1.0f - ALPHA) * a * ni[i] + ALPHA * x0[idx];
    agg[idx] = 0.f;
}

// K5: GIN0 scatter: agg[dst] += h[src]  (no normalization)
__global__ void k_scatter128_plain(const float* __restrict__ h,
                                   const int* __restrict__ src,
                                   const int* __restrict__ dst,
                                   float* __restrict__ agg, int E) {
    size_t t = (size_t)blockIdx.x * blockDim.x + threadIdx.x;
    int e = (int)(t >> 5);
    if (e >= E) return;
    int lane = (int)(t & 31);
    int s = src[e], d = dst[e];
    const float4 v = *reinterpret_cast<const float4*>(h + (size_t)s * D128 + lane * 4);
    float* ap = agg + (size_t)d * D128 + lane * 4;
    atomAdd(ap + 0, v.x);
    atomAdd(ap + 1, v.y);
    atomAdd(ap + 2, v.z);
    atomAdd(ap + 3, v.w);
}

// ---------------------------------------------------------------------------
// K6: GIN0 GEMM: out1 = ((1+eps1)*h + agg128) @ W1 + b1
// fp32 WMMA 16x16x4, one wave per 16x16 output tile, K=128.
// f32 A layout: per lane v2f = A(m, k+2*half), A(m, k+2*half+1), m = lane&15
// f32 B layout: per lane v2f = B(k+2*half, n), B(k+2*half+1, n),  n = lane&15
// C/D: 8 VGPRs, element g -> (M = g + 8*half, N = lane&15)
// ---------------------------------------------------------------------------
__global__ void k_gemm_gin0(const float* __restrict__ h, const float* __restrict__ agg,
                            const float* __restrict__ W, const float* __restrict__ bias,
                            const float* __restrict__ eps, float* __restrict__ out,
                            int tilesM) {
    int waveId = blockIdx.x * (blockDim.x >> 5) + (threadIdx.x >> 5);
    int tm = waveId >> 3;            // 8 column tiles (128/16)
    int tn = waveId & 7;
    if (tm >= tilesM) return;        // wave-uniform: EXEC stays all-1 for WMMA
    int lane = threadIdx.x & 31;
    int half = lane >> 4, lid = lane & 15;
    int m   = tm * 16 + lid;
    int col = tn * 16 + lid;
    float e1 = 1.0f + eps[0];
    const float* hrow = h   + (size_t)m * D128;
    const float* arow = agg + (size_t)m * D128;
    v8f c = {};
    for (int k = 0; k < D128; k += 4) {
        int kk = k + 2 * half;
        v2f a, b;
        a.x = fmaf(e1, hrow[kk],     arow[kk]);
        a.y = fmaf(e1, hrow[kk + 1], arow[kk + 1]);
        b.x = W[(size_t)kk * D128 + col];
        b.y = W[(size_t)(kk + 1) * D128 + col];
        c = __builtin_amdgcn_wmma_f32_16x16x4_f32(false, a, false, b,
                                                  (short)0, c, false, false);
    }
    float bv = bias[col];
    for (int g = 0; g < 8; ++g) {
        int row = tm * 16 + g + 8 * half;
        out[(size_t)row * D128 + col] = c[g] + bv;
    }
}

// ---------------------------------------------------------------------------
// K7: GIN1 scatter on virtual concat [out1 | x0]: agg256[dst] += concat[src]
// one wave per edge, 8 floats per lane; lanes 0-15 read out1, 16-31 read x0
// ---------------------------------------------------------------------------
__global__ void k_scatter256_concat(const float* __restrict__ out1,
                                    const float* __restrict__ x0,
                                    const int* __restrict__ src,
                                    const int* __restrict__ dst,
                                    float* __restrict__ agg256, int E) {
    size_t t = (size_t)blockIdx.x * blockDim.x + threadIdx.x;
    int e = (int)(t >> 5);
    if (e >= E) return;
    int lane = (int)(t & 31);
    int fb = lane * 8;               // feature base in [0,256)
    int s = src[e], d = dst[e];
    const float* sp = (fb < D128) ? (out1 + (size_t)s * D128 + fb)
                                  : (x0   + (size_t)s * D128 + fb - D128);
    float4 v0 = *reinterpret_cast<const float4*>(sp);
    float4 v1 = *reinterpret_cast<const float4*>(sp + 4);
    float* ap = agg256 + (size_t)d * 2 * D128 + fb;
    atomAdd(ap + 0, v0.x); atomAdd(ap + 1, v0.y);
    atomAdd(ap + 2, v0.z); atomAdd(ap + 3, v0.w);
    atomAdd(ap + 4, v1.x); atomAdd(ap + 5, v1.y);
    atomAdd(ap + 6, v1.z); atomAdd(ap + 7, v1.w);
}

// ---------------------------------------------------------------------------
// K8: GIN1 GEMM + Mish: out = mish( ((1+eps2)*[out1|x0] + agg256) @ W2 + b2 )
// K=256; concat handled by pointer select on the A-fragment source.
// ---------------------------------------------------------------------------
__global__ void k_gemm_gin1_mish(const float* __restrict__ out1,
                                 const float* __restrict__ x0,
                                 const float* __restrict__ agg256,
                                 const float* __restrict__ W,
                                 const float* __restrict__ bias,
                                 const float* __restrict__ eps,
                                 float* __restrict__ out, int tilesM) {
    int waveId = blockIdx.x * (blockDim.x >> 5) + (threadIdx.x >> 5);
    int tm = waveId >> 3;
    int tn = waveId & 7;
    if (tm >= tilesM) return;
    int lane = threadIdx.x & 31;
    int half = lane >> 4, lid = lane & 15;
    int m   = tm * 16 + lid;
    int col = tn * 16 + lid;
    float e2 = 1.0f + eps[0];
    const float* o1row = out1   + (size_t)m * D128;
    const float* x0row = x0     + (size_t)m * D128;
    const float* a2row = agg256 + (size_t)m * 2 * D128;
    v8f c = {};
    for (int k = 0; k < 2 * D128; k += 4) {
        int kk = k + 2 * half;       // kk even; kk and kk+1 on same side of 128
        const float* sp = (kk < D128) ? (o1row + kk) : (x0row + kk - D128);
        v2f a, b;
        a.x = fmaf(e2, sp[0], a2row[kk]);
        a.y = fmaf(e2, sp[1], a2row[kk + 1]);
        b.x = W[(size_t)kk * D128 + col];
        b.y = W[(size_t)(kk + 1) * D128 + col];
        c = __builtin_amdgcn_wmma_f32_16x16x4_f32(false, a, false, b,
                                                  (short)0, c, false, false);
    }
    float bv = bias[col];
    for (int g = 0; g < 8; ++g) {
        int row = tm * 16 + g + 8 * half;
        float v = c[g] + bv;
        float sp = (v > 20.0f) ? v : log1pf(expf(v));   // softplus
        out[(size_t)row * D128 + col] = v * tanhf(sp);  // mish
    }
}

// ---------------------------------------------------------------------------
// Host-side orchestration
// ---------------------------------------------------------------------------
extern "C" void kernel_launch(void* const* d_in, const int* in_sizes, int n_in,
                              void* d_out, int out_size, void* d_ws, size_t ws_size,
                              hipStream_t stream) {
    const float* x    = (const float*)d_in[0];
    const int*   src  = (const int*)  d_in[1];
    const int*   dst  = (const int*)  d_in[2];
    const float* W1   = (const float*)d_in[3];
    const float* b1   = (const float*)d_in[4];
    const float* W2   = (const float*)d_in[5];
    const float* b2   = (const float*)d_in[6];
    const float* eps1 = (const float*)d_in[7];
    const float* eps2 = (const float*)d_in[8];

    const int N = in_sizes[0] / D128;   // 50000
    const int E = in_sizes[1];          // 600000

    // workspace layout (floats): x0 | h | agg128 | out1 | agg256 | norm_o | norm_i
    float* ws      = (float*)d_ws;
    size_t n128    = (size_t)N * D128;
    float* x0      = ws;
    float* h       = x0     + n128;
    float* agg128  = h      + n128;
    float* out1    = agg128 + n128;
    float* agg256  = out1   + n128;
    float* norm_o  = agg256 + (size_t)N * 2 * D128;
    float* norm_i  = norm_o + N;

    const int BLK = 256;
    float* outp = (float*)d_out;

    // init: tanh + zero scratch
    {
        size_t tot = (size_t)N * 2 * D128;
        int blocks = (int)((tot + BLK - 1) / BLK);
        k_init<<<blocks, BLK, 0, stream>>>(x, x0, h, agg128, agg256, norm_o, norm_i, N);
    }
    // degrees -> norms
    k_degrees<<<(E + BLK - 1) / BLK, BLK, 0, stream>>>(src, dst, norm_o, norm_i, E);
    k_norm_fin<<<(N + BLK - 1) / BLK, BLK, 0, stream>>>(norm_o, norm_i, N);

    // APPNP: 5 power iterations
    {
        size_t tE = (size_t)E * 32;
        int eBlocks = (int)((tE + BLK - 1) / BLK);
        int uBlocks = (int)(((size_t)N * D128 + BLK - 1) / BLK);
        for (int it = 0; it < 5; ++it) {
            k_scatter128_scaled<<<eBlocks, BLK, 0, stream>>>(h, norm_o, src, dst, agg128, E);
            k_appnp_update<<<uBlocks, BLK, 0, stream>>>(agg128, x0, norm_i, h, N);
        }
        // GIN0 aggregation (agg128 re-zeroed by last update)
        k_scatter128_plain<<<eBlocks, BLK, 0, stream>>>(h, src, dst, agg128, E);
    }

    // GIN0 GEMM (WMMA fp32): out1 = ((1+eps1)h + agg) @ W1 + b1
    {
        int tilesM = N / 16;                 // 3125, exact
        int waves  = tilesM * (D128 / 16);   // 25000
        int blocks = (waves + 7) / 8;        // 8 waves / block
        k_gemm_gin0<<<blocks, BLK, 0, stream>>>(h, agg128, W1, b1, eps1, out1, tilesM);
    }

    // GIN1 aggregation on virtual concat [out1 | x0]
    {
        size_t tE = (size_t)E * 32;
        int eBlocks = (int)((tE + BLK - 1) / BLK);
        k_scatter256_concat<<<eBlocks, BLK, 0, stream>>>(out1, x0, src, dst, agg256, E);
    }

    // GIN1 GEMM (WMMA fp32, K=256) + bias + Mish -> d_out
    {
        int tilesM = N / 16;
        int waves  = tilesM * (D128 / 16);
        int blocks = (waves + 7) / 8;
        k_gemm_gin1_mish<<<blocks, BLK, 0, stream>>>(out1, x0, agg256, W2, b2, eps2,
                                                     outp, tilesM);
    }
}